// GraphEmbedding_29506425324286
// MI455X (gfx1250) — compile-verified
//
#include <hip/hip_runtime.h>

// ---------------------------------------------------------------------------
// 2-layer GraphConv (DGL norm='both') for MI455X / gfx1250.
// Memory-bound problem (~1 GB traffic, ~45-90us floor @ 23.3 TB/s).
// Dense transforms use V_WMMA_F32_16X16X4_F32 (the fp32 CDNA5 WMMA shape).
// W stored TRANSPOSED in LDS -> A and B fragments are single ds_load_b64s.
// Full blocks take an unconditional store path (no per-row EXEC masking).
// ---------------------------------------------------------------------------

typedef __attribute__((ext_vector_type(2))) float v2f;
typedef __attribute__((ext_vector_type(8))) float v8f;

// ---- zero a float buffer (grid-stride) ------------------------------------
__global__ void k_zero(float* __restrict__ p, int n) {
    int i = blockIdx.x * blockDim.x + threadIdx.x;
    int stride = gridDim.x * blockDim.x;
    for (; i < n; i += stride) p[i] = 0.0f;
}

// ---- degree accumulation (float atomics; exact for counts < 2^24) ---------
__global__ void k_degree(const int* __restrict__ src, const int* __restrict__ dst,
                         float* __restrict__ deg_out, float* __restrict__ deg_in, int E) {
    int e = blockIdx.x * blockDim.x + threadIdx.x;
    if (e < E) {
        atomicAdd(&deg_out[src[e]], 1.0f);
        atomicAdd(&deg_in[dst[e]], 1.0f);
    }
}

// ---- deg -> rsqrt(max(deg,1)) in place ------------------------------------
__global__ void k_norm(float* __restrict__ deg, int n) {
    int i = blockIdx.x * blockDim.x + threadIdx.x;
    if (i < n) {
        float d = deg[i];
        deg[i] = rsqrtf(d < 1.0f ? 1.0f : d);
    }
}

// ---- H = (X * norm_row) @ W   where X:[N,32], W:[32,32] -------------------
// 256 threads = 8 waves; block covers 128 rows; each wave: 16 rows x 32 cols
// as two 16x16 tiles, K=32 via 8 chained V_WMMA_F32_16X16X4_F32 per tile.
__global__ __launch_bounds__(256)
void k_gemm32_wmma(const float* __restrict__ X, const float* __restrict__ norm,
                   const float* __restrict__ W, float* __restrict__ H, int N) {
    __shared__ float Wp[32][34];     // W TRANSPOSED: Wp[j][k] = W[k][j]; stride 34
    __shared__ float Xs[128][34];    // stride 34: conflict-free, 8B aligned pairs

    const int tid = threadIdx.x;
    const int blockRow = blockIdx.x * 128;

    // Load W (1024 floats) cooperatively, transposing into Wp[j][k].
#pragma unroll
    for (int i = 0; i < 4; ++i) {
        int idx = tid + i * 256;          // idx = k*32 + j
        Wp[idx & 31][idx >> 5] = W[idx];
    }
    // Load X tile (128x32) scaled by norm_src; zero-pad out-of-range rows.
#pragma unroll
    for (int i = 0; i < 4; ++i) {
        int idx = tid + i * 256;          // float4 index 0..1023
        int r   = idx >> 3;
        int c4  = (idx & 7) << 2;
        int gr  = blockRow + r;
        float4 v = make_float4(0.f, 0.f, 0.f, 0.f);
        float  s = 0.f;
        if (gr < N) {
            v = *(const float4*)(X + (size_t)gr * 32 + c4);
            s = norm[gr];
        }
        Xs[r][c4 + 0] = v.x * s;
        Xs[r][c4 + 1] = v.y * s;
        Xs[r][c4 + 2] = v.z * s;
        Xs[r][c4 + 3] = v.w * s;
    }
    __syncthreads();

    const int wave = tid >> 5;
    const int lane = tid & 31;
    const int half = lane >> 4;       // 0: lanes 0-15, 1: lanes 16-31
    const int l16  = lane & 15;

    const float* xr  = &Xs[wave * 16 + l16][0];
    const float* wr0 = &Wp[l16][0];        // tile0 column
    const float* wr1 = &Wp[l16 + 16][0];   // tile1 column

    v8f acc0 = {};
    v8f acc1 = {};
#pragma unroll
    for (int k = 0; k < 8; ++k) {
        // A 16x4 fragment (ISA layout): lanes0-15 carry K=4k+0,4k+1 of row l16;
        // lanes16-31 carry K=4k+2,4k+3.  Single ds_load_b64.
        int kb = 4 * k + 2 * half;
        v2f a  = *(const v2f*)(xr + kb);
        // B 4x16 fragment: VGPR0 = rows K0 (lanes0-15) / K2 (lanes16-31),
        // VGPR1 = rows K1 / K3. Adjacent in k in Wp -> single ds_load_b64.
        v2f b0 = *(const v2f*)(wr0 + kb);
        v2f b1 = *(const v2f*)(wr1 + kb);
        acc0 = __builtin_amdgcn_wmma_f32_16x16x4_f32(false, a, false, b0,
                                                     (short)0, acc0, false, false);
        acc1 = __builtin_amdgcn_wmma_f32_16x16x4_f32(false, a, false, b1,
                                                     (short)0, acc1, false, false);
    }

    // C/D layout: VGPR r -> row (r + 8*half), col = l16 (+16 for tile1).
    const int rowBase = blockRow + wave * 16 + half * 8;
    if (blockRow + 128 <= N) {
        // Fast path (all but the last block): unconditional, clause-able stores.
#pragma unroll
        for (int r = 0; r < 8; ++r) {
            float* hp = H + (size_t)(rowBase + r) * 32 + l16;
            hp[0]  = acc0[r];
            hp[16] = acc1[r];
        }
    } else {
#pragma unroll
        for (int r = 0; r < 8; ++r) {
            int row = rowBase + r;
            if (row < N) {
                H[(size_t)row * 32 + l16]      = acc0[r];
                H[(size_t)row * 32 + 16 + l16] = acc1[r];
            }
        }
    }
}

// ---- agg[dst] += H[src] ; 8 lanes per edge, float4 each -------------------
__global__ void k_scatter(const float* __restrict__ H, const int* __restrict__ src,
                          const int* __restrict__ dst, float* __restrict__ A, int E) {
    int t = blockIdx.x * blockDim.x + threadIdx.x;
    int e = t >> 3;
    if (e >= E) return;
    int c = (t & 7) << 2;
    int s = src[e];
    int d = dst[e];
    float4 v = *(const float4*)(H + (size_t)s * 32 + c);   // 128B coalesced gather
    float* out = A + (size_t)d * 32 + c;
    atomicAdd(out + 0, v.x);
    atomicAdd(out + 1, v.y);
    atomicAdd(out + 2, v.z);
    atomicAdd(out + 3, v.w);
}

// ---- Y = (relu?)(A * norm_dst + bias) -------------------------------------
__global__ void k_finalize(const float* __restrict__ A, const float* __restrict__ ndst,
                           const float* __restrict__ bias, float* __restrict__ Y,
                           int N, int do_relu) {
    int t = blockIdx.x * blockDim.x + threadIdx.x;
    int row = t >> 3;
    if (row >= N) return;
    int c = (t & 7) << 2;
    float s = ndst[row];
    float4 v = *(const float4*)(A + (size_t)row * 32 + c);
    float4 b = *(const float4*)(bias + c);
    float4 o;
    o.x = v.x * s + b.x;
    o.y = v.y * s + b.y;
    o.z = v.z * s + b.z;
    o.w = v.w * s + b.w;
    if (do_relu) {
        o.x = fmaxf(o.x, 0.f); o.y = fmaxf(o.y, 0.f);
        o.z = fmaxf(o.z, 0.f); o.w = fmaxf(o.w, 0.f);
    }
    *(float4*)(Y + (size_t)row * 32 + c) = o;
}

extern "C" void kernel_launch(void* const* d_in, const int* in_sizes, int n_in,
                              void* d_out, int out_size, void* d_ws, size_t ws_size,
                              hipStream_t stream) {
    const float* X   = (const float*)d_in[0];
    const int*   src = (const int*)d_in[1];
    const int*   dst = (const int*)d_in[2];
    const float* W1  = (const float*)d_in[3];
    const float* b1  = (const float*)d_in[4];
    const float* W2  = (const float*)d_in[5];
    const float* b2  = (const float*)d_in[6];
    float* out = (float*)d_out;

    const int N = in_sizes[0] / 32;
    const int E = in_sizes[1];

    // Workspace layout (floats): [norm_src N][norm_dst N][H 32N][A 32N][Y 32N]
    float* w    = (float*)d_ws;
    float* nsrc = w;
    float* ndst = w + (size_t)N;
    float* H    = w + 2 * (size_t)N;
    float* A    = H + 32 * (size_t)N;
    float* Y    = A + 32 * (size_t)N;

    const int TB = 256;
    const int gridE    = (E + TB - 1) / TB;          // per-edge
    const int gridE8   = (8 * E + TB - 1) / TB;      // 8 lanes per edge
    const int gridN8   = (8 * N + TB - 1) / TB;      // 8 lanes per row
    const int gridGemm = (N + 127) / 128;

    // Degrees -> norms (shared by both layers). Re-zeroed every call.
    k_zero<<<512, TB, 0, stream>>>(w, 2 * N);
    k_degree<<<gridE, TB, 0, stream>>>(src, dst, nsrc, ndst, E);
    k_norm<<<(2 * N + TB - 1) / TB, TB, 0, stream>>>(w, 2 * N);

    // ---- Layer 1 ----
    k_gemm32_wmma<<<gridGemm, TB, 0, stream>>>(X, nsrc, W1, H, N);
    k_zero<<<1024, TB, 0, stream>>>(A, 32 * N);
    k_scatter<<<gridE8, TB, 0, stream>>>(H, src, dst, A, E);
    k_finalize<<<gridN8, TB, 0, stream>>>(A, ndst, b1, Y, N, /*relu=*/1);

    // ---- Layer 2 ----
    k_gemm32_wmma<<<gridGemm, TB, 0, stream>>>(Y, nsrc, W2, H, N);
    k_zero<<<1024, TB, 0, stream>>>(A, 32 * N);
    k_scatter<<<gridE8, TB, 0, stream>>>(H, src, dst, A, E);
    k_finalize<<<gridN8, TB, 0, stream>>>(A, ndst, b2, out, N, /*relu=*/0);
}